// ManualRNN_90898687852956
// MI455X (gfx1250) — compile-verified
//
#include <hip/hip_runtime.h>
#include <math.h>

// RNN: B=32, T=2048, Fin=H=O=256
#define RNN_B   32
#define RNN_T   2048
#define RNN_K   256        // Fin == H == O == 256
#define LPITCH  260        // LDS row pitch (floats): 260 % 64 = 4 banks/lane -> conflict-free

typedef __attribute__((ext_vector_type(2))) float v2f;
typedef __attribute__((ext_vector_type(8))) float v8f;

// D = A(16x4 f32) x B(4x16 f32) + C(16x16 f32)
#define WMMA_F32(a, b, c) \
    __builtin_amdgcn_wmma_f32_16x16x4_f32(false, (a), false, (b), (short)0, (c), false, false)

// ---------------------------------------------------------------------------
// Kernel A: xh = x @ Wxh^T + bxh.  x is (B,T,Fin) row-major => M = B*T rows.
// Output written to zh in (T, B, H) order so the scan reads contiguous tiles.
// One block = 16 rows (one m-tile) x 16 waves (one n-tile each), K = 256.
// ---------------------------------------------------------------------------
__global__ __launch_bounds__(512) void rnn_xproj(const float* __restrict__ x,
                                                 const float* __restrict__ Wxh,
                                                 const float* __restrict__ bxh,
                                                 float* __restrict__ zh)
{
    __shared__ float xs[16 * RNN_K];          // 16KB staged A tile
    const int r0 = blockIdx.x * 16;           // row block in (b,t) flat space

    for (int i = threadIdx.x; i < 16 * RNN_K; i += 512)
        xs[i] = x[(size_t)r0 * RNN_K + i];
    __syncthreads();

    const int lane = threadIdx.x & 31;
    const int l15  = lane & 15;
    const int lh   = lane >> 4;               // half-wave select (K+0/1 vs K+2/3)
    const int col0 = (threadIdx.x >> 5) * 16; // n-tile per wave

    const float* __restrict__ wrow = Wxh + (size_t)(col0 + l15) * RNN_K;
    const float* __restrict__ arow = xs + l15 * RNN_K;

    v8f c0 = {}, c1 = {};
    #pragma unroll 4
    for (int k = 0; k < RNN_K; k += 8) {
        const int ka = k + 2 * lh;
        const int kb = ka + 4;
        v2f a0, b0, a1, b1;
        a0.x = arow[ka];     a0.y = arow[ka + 1];
        b0.x = wrow[ka];     b0.y = wrow[ka + 1];
        a1.x = arow[kb];     a1.y = arow[kb + 1];
        b1.x = wrow[kb];     b1.y = wrow[kb + 1];
        c0 = WMMA_F32(a0, b0, c0);
        c1 = WMMA_F32(a1, b1, c1);
    }

    const float bias = bxh[col0 + l15];
    #pragma unroll
    for (int r = 0; r < 8; ++r) {
        const int m   = r + 8 * lh;           // C/D layout: VGPR r -> rows r, r+8
        const int row = r0 + m;               // flat (b,t): b = row/T, t = row%T
        const int b   = row >> 11;            // /2048
        const int t   = row & (RNN_T - 1);
        zh[((size_t)t * RNN_B + b) * RNN_K + col0 + l15] = c0[r] + c1[r] + bias;
    }
}

// ---------------------------------------------------------------------------
// Kernel B: sequential scan  h = tanh(xh_t + h @ Whh^T).
// Batch rows are independent -> 2 blocks of 16 batch rows each (M = 16); each
// block owns a full WGP, so the entire 256KB Whh is staged into the 320KB LDS
// once, padded to a 260-float pitch (bank-conflict-free fragment reads). The
// 2048-step critical loop is then pure ds_load -> v_wmma.
// zh[t] (the xh tile) is overwritten in place with h[t].
// ---------------------------------------------------------------------------
__global__ __launch_bounds__(512) void rnn_scan(const float* __restrict__ h0,
                                                const float* __restrict__ Whh,
                                                float* __restrict__ zh,
                                                float* __restrict__ hfinal)
{
    extern __shared__ float smem[];
    float* __restrict__ hsm = smem;                    // 16  x LPITCH: hidden state
    float* __restrict__ wsm = smem + 16 * LPITCH;      // 256 x LPITCH: Whh (row-major, padded)

    const int bbase = blockIdx.x * 16;                 // batch-row base for this block

    for (int i = threadIdx.x; i < 16 * RNN_K; i += 512)
        hsm[(i >> 8) * LPITCH + (i & 255)] = h0[(size_t)bbase * RNN_K + i];
    for (int i = threadIdx.x; i < RNN_K * RNN_K; i += 512)
        wsm[(i >> 8) * LPITCH + (i & 255)] = Whh[i];
    __syncthreads();

    const int lane = threadIdx.x & 31;
    const int l15  = lane & 15;
    const int lh   = lane >> 4;
    const int col0 = (threadIdx.x >> 5) * 16;

    const float* __restrict__ wrow = wsm + (col0 + l15) * LPITCH;
    const float* __restrict__ arow = hsm + l15 * LPITCH;

    for (int t = 0; t < RNN_T; ++t) {
        if (t + 1 < RNN_T) {                  // warm next xh tile (global_prefetch_b8)
            const float* nxt =
                zh + ((size_t)(t + 1) * RNN_B + bbase) * RNN_K + threadIdx.x * 8;
            __builtin_prefetch(nxt, 0, 0);
        }

        v8f c0 = {}, c1 = {};
        #pragma unroll 4
        for (int k = 0; k < RNN_K; k += 8) {
            const int ka = k + 2 * lh;
            const int kb = ka + 4;
            v2f a0, b0, a1, b1;
            a0.x = arow[ka];     a0.y = arow[ka + 1];
            b0.x = wrow[ka];     b0.y = wrow[ka + 1];
            a1.x = arow[kb];     a1.y = arow[kb + 1];
            b1.x = wrow[kb];     b1.y = wrow[kb + 1];
            c0 = WMMA_F32(a0, b0, c0);
            c1 = WMMA_F32(a1, b1, c1);
        }

        const size_t zrow = (size_t)t * RNN_B + bbase;
        float hn[8];
        #pragma unroll
        for (int r = 0; r < 8; ++r) {
            const int m = r + 8 * lh;
            const float xv = zh[(zrow + m) * RNN_K + col0 + l15];
            hn[r] = tanhf(c0[r] + c1[r] + xv);
        }
        __syncthreads();                      // all step-t reads of hsm done
        #pragma unroll
        for (int r = 0; r < 8; ++r) {
            const int m = r + 8 * lh;
            hsm[m * LPITCH + col0 + l15] = hn[r];
            zh[(zrow + m) * RNN_K + col0 + l15] = hn[r];   // hs[t] for output GEMM
        }
        __syncthreads();                      // h_next visible before step t+1
    }

    #pragma unroll
    for (int r = 0; r < 8; ++r) {             // h_final -> tail of d_out
        const int m = r + 8 * lh;
        hfinal[(size_t)(bbase + m) * RNN_K + col0 + l15] = hsm[m * LPITCH + col0 + l15];
    }
}

// ---------------------------------------------------------------------------
// Kernel C: y = hs @ Why^T + by.  hs (=zh) is (T,B,H); y is (B,T,O).
// ---------------------------------------------------------------------------
__global__ __launch_bounds__(512) void rnn_yproj(const float* __restrict__ zh,
                                                 const float* __restrict__ Why,
                                                 const float* __restrict__ by,
                                                 float* __restrict__ y)
{
    __shared__ float hx[16 * RNN_K];
    const int r0 = blockIdx.x * 16;           // row block in (t,b) flat space

    for (int i = threadIdx.x; i < 16 * RNN_K; i += 512)
        hx[i] = zh[(size_t)r0 * RNN_K + i];
    __syncthreads();

    const int lane = threadIdx.x & 31;
    const int l15  = lane & 15;
    const int lh   = lane >> 4;
    const int col0 = (threadIdx.x >> 5) * 16;

    const float* __restrict__ wrow = Why + (size_t)(col0 + l15) * RNN_K;
    const float* __restrict__ arow = hx + l15 * RNN_K;

    v8f c0 = {}, c1 = {};
    #pragma unroll 4
    for (int k = 0; k < RNN_K; k += 8) {
        const int ka = k + 2 * lh;
        const int kb = ka + 4;
        v2f a0, b0, a1, b1;
        a0.x = arow[ka];     a0.y = arow[ka + 1];
        b0.x = wrow[ka];     b0.y = wrow[ka + 1];
        a1.x = arow[kb];     a1.y = arow[kb + 1];
        b1.x = wrow[kb];     b1.y = wrow[kb + 1];
        c0 = WMMA_F32(a0, b0, c0);
        c1 = WMMA_F32(a1, b1, c1);
    }

    const float bias = by[col0 + l15];
    #pragma unroll
    for (int r = 0; r < 8; ++r) {
        const int m   = r + 8 * lh;
        const int row = r0 + m;               // (t,b): t = row/32, b = row%32
        const int t   = row >> 5;
        const int b   = row & (RNN_B - 1);
        y[((size_t)b * RNN_T + t) * RNN_K + col0 + l15] = c0[r] + c1[r] + bias;
    }
}

// ---------------------------------------------------------------------------
extern "C" void kernel_launch(void* const* d_in, const int* in_sizes, int n_in,
                              void* d_out, int out_size, void* d_ws, size_t ws_size,
                              hipStream_t stream) {
    (void)in_sizes; (void)n_in; (void)out_size; (void)ws_size;

    const float* x   = (const float*)d_in[0];   // (B,T,Fin)
    const float* h0  = (const float*)d_in[1];   // (B,H)
    const float* Wxh = (const float*)d_in[2];   // (H,Fin)
    const float* bxh = (const float*)d_in[3];   // (H)
    const float* Whh = (const float*)d_in[4];   // (H,H)
    const float* Why = (const float*)d_in[5];   // (O,H)
    const float* by  = (const float*)d_in[6];   // (O)

    float* y      = (float*)d_out;                                  // (B,T,O)
    float* hfinal = y + (size_t)RNN_B * RNN_T * RNN_K;              // (B,H) tail
    float* zh     = (float*)d_ws;   // (T,B,H) f32 = 64 MB: xh then hs in place

    const int mtiles = (RNN_B * RNN_T) / 16;    // 4096
    const size_t scan_lds = (size_t)(16 + RNN_K) * LPITCH * sizeof(float); // ~283 KB of 320 KB

    rnn_xproj<<<mtiles, 512, 0, stream>>>(x, Wxh, bxh, zh);
    rnn_scan <<<RNN_B / 16, 512, scan_lds, stream>>>(h0, Whh, zh, hfinal);
    rnn_yproj<<<mtiles, 512, 0, stream>>>(zh, Why, by, y);
}